// PickupCPW_87746181857532
// MI455X (gfx1250) — compile-verified
//
#include <hip/hip_runtime.h>
#include <math.h>

// ---------- problem constants ----------
#define NXc 192
#define NYc 192
#define NTc 8
#define NK  (NXc * NYc)      // 36864 flattened (a,b)
#define NP  12
#define PJc (NP * NYc)       // 2304 flattened (p,j)
#define NTILES (PJc / 16)    // 144 output column tiles

static constexpr float DXf  = 5e-8f;
static constexpr float DYf  = 5e-8f;
static constexpr float DZf  = 5e-8f;
static constexpr float DTf  = 1e-11f;
static constexpr float Ff   = 6.0e9f;
static constexpr float ZPf  = 1e-7f;
static constexpr float INV_DX  = 1.0f / 5e-8f;   // 2e7
static constexpr float INV_2DX = 0.5f / 5e-8f;   // 1e7
static constexpr float KVf  = 1e-7f * (5e-8f * 5e-8f * 5e-8f); // mu0/4pi * dx*dy*dz
static constexpr float KXf  = 1e-7f * (5e-8f * 5e-8f);         // mu0/4pi * dy*dz
static constexpr float PIf  = 3.14159265358979323846f;

__constant__ int   c_xpos[NP] = {16, 30, 44, 58, 72, 86, 100, 114, 128, 142, 156, 170};
__constant__ float c_w[NP]    = {-0.25f, -0.25f, -0.25f, -0.25f,
                                  0.25f,  0.25f,  0.25f,  0.25f,
                                 -0.25f, -0.25f, -0.25f, -0.25f};

typedef __attribute__((ext_vector_type(2))) float v2f;
typedef __attribute__((ext_vector_type(8))) float v8f;

// ---------------------------------------------------------------------------
// Kernel 1: W[t][k] = Kv*(-Msat*dmz/dx) folded with the x-boundary cross terms.
// W has 16 t-rows; rows 8..15 are zero padding so the WMMA A-operand loads are
// uniform across all 32 lanes (EXEC must stay all-1s for WMMA).
// ---------------------------------------------------------------------------
__global__ void __launch_bounds__(256) prep_W(const float* __restrict__ m,
                                              const float* __restrict__ msat,
                                              float* __restrict__ W) {
    int gid = blockIdx.x * blockDim.x + threadIdx.x;
    if (gid >= 16 * NK) return;
    int t = gid / NK;
    if (t >= NTc) { W[gid] = 0.0f; return; }
    int k = gid - t * NK;
    int a = k / NYc;
    int b = k - a * NYc;
    const float Ms = msat[0];
    const float* mz = m + (t * 3 + 2) * NK;   // m[(t,2,:,:)]
    float d;
    if (a == 0)            d = (mz[NYc + b]              - mz[b])                * INV_DX;
    else if (a == NXc - 1) d = (mz[(NXc - 1) * NYc + b]  - mz[(NXc - 2) * NYc + b]) * INV_DX;
    else                   d = (mz[(a + 1) * NYc + b]    - mz[(a - 1) * NYc + b]) * INV_2DX;
    float w = -KVf * Ms * d;                       // Kv * curl_y (only slice used)
    if (a == NXc - 1) w += KXf * mz[(NXc - 1) * NYc + b]; // C_100[:,1] @ invd[:,:,NX-1,:]
    if (a == 1)       w -= KXf * mz[b];                   // C_n00[:,1] @ invd[:,:,1,:]
    W[gid] = w;
}

// ---------------------------------------------------------------------------
// Kernel 2: A1[t][pj] = sum_k W[t][k] * invd[pj][k]  via V_WMMA_F32_16X16X4_F32.
// One block = 16 pj-columns. 4 waves split the K=36864 reduction (48 a-rows
// each); invd is generated per-lane in VALU (co-executes with the XDL pipe),
// partial 16x16 f32 accumulators are combined through LDS.
// ---------------------------------------------------------------------------
__global__ void __launch_bounds__(128) gemm_A1(const float* __restrict__ W,
                                               float* __restrict__ A1) {
    __shared__ float red[4][8][16];

    const int lane = threadIdx.x & 31;
    const int wave = threadIdx.x >> 5;
    const int half = lane >> 4;        // K sub-pair: 0 -> K{0,1}, 1 -> K{2,3}
    const int nc   = lane & 15;        // A row (t) and B column (pj) index

    const int pj = blockIdx.x * 16 + nc;
    const int p  = pj / NYc;
    const int j  = pj - p * NYc;
    const float xp  = (float)c_xpos[p] * DXf;
    const float z2  = ZPf * ZPf;
    const float fj  = (float)j;

    const float* __restrict__ wrow = W + nc * NK;  // A operand row t = nc (rows 8..15 are zeros)

    v8f acc = {};  // C = 0

    const int a0 = wave * 48;
    for (int a = a0; a < a0 + 48; ++a) {
        const float dx  = (float)a * DXf - xp;
        const float dxz = __builtin_fmaf(dx, dx, z2);
        const float* __restrict__ pA = wrow + a * NYc + 2 * half;
        __builtin_prefetch(pA + NYc, 0, 0);        // next a-row -> global_prefetch
        #pragma unroll 4
        for (int bi = 0; bi < NYc; bi += 4) {
            const int kb = bi + 2 * half;
            // B operand: invd(p,j; a, kb) and invd(p,j; a, kb+1)
            const float db0 = ((float)kb - fj) * DYf;
            const float db1 = ((float)(kb + 1) - fj) * DYf;
            v2f bv;
            bv.x = rsqrtf(__builtin_fmaf(db0, db0, dxz));
            bv.y = rsqrtf(__builtin_fmaf(db1, db1, dxz));
            // A operand: W[t=nc][a*192 + kb + {0,1}] (8B-aligned, uniform EXEC)
            const v2f av = *(const v2f*)(pA + bi);
            // D(16x16,f32) += A(16x4,f32) x B(4x16,f32)
            acc = __builtin_amdgcn_wmma_f32_16x16x4_f32(
                false, av, false, bv, (short)0, acc, false, false);
        }
    }

    // D layout: VGPR r, lanes 0-15 hold row t=r (rows 8..15 are exact zeros).
    if (lane < 16) {
        #pragma unroll
        for (int r = 0; r < 8; ++r) red[wave][r][lane] = acc[r];
    }
    __syncthreads();
    if (threadIdx.x < 128) {
        const int r = threadIdx.x >> 4;
        const int c = threadIdx.x & 15;
        const float s = red[0][r][c] + red[1][r][c] + red[2][r][c] + red[3][r][c];
        A1[r * PJc + blockIdx.x * 16 + c] = s;
    }
}

// ---------------------------------------------------------------------------
// Kernel 3: finite difference in t, Fourier projection, CPW weighting -> scalar.
// (zr==re, zi==im identically, so only re/im are needed.)
// ---------------------------------------------------------------------------
__global__ void __launch_bounds__(256) finalize(const float* __restrict__ A1,
                                                float* __restrict__ out) {
    __shared__ float sred[2];
    if (threadIdx.x == 0) { sred[0] = 0.0f; sred[1] = 0.0f; }
    __syncthreads();

    float cosv[NTc - 1], sinv[NTc - 1];
    #pragma unroll
    for (int tau = 0; tau < NTc - 1; ++tau) {
        const float tt  = (float)(1000 - NTc + 1 + tau) * DTf;   // (993+tau)*DT
        const float ang = 2.0f * PIf * Ff * tt;
        cosv[tau] = cosf(ang);
        sinv[tau] = sinf(ang);
    }

    float sr = 0.0f, si = 0.0f;
    for (int pj = threadIdx.x; pj < PJc; pj += 256) {
        const int p = pj / NYc;
        const float wp = c_w[p];
        float prev = A1[pj];
        float ac = 0.0f, as = 0.0f;
        #pragma unroll
        for (int tau = 0; tau < NTc - 1; ++tau) {
            const float cur = A1[(tau + 1) * PJc + pj];
            const float dv  = cur - prev;        // deltaV term (scale applied later)
            ac = __builtin_fmaf(dv, cosv[tau], ac);
            as = __builtin_fmaf(dv, sinv[tau], as);
            prev = cur;
        }
        sr = __builtin_fmaf(wp, ac, sr);
        si = __builtin_fmaf(wp, as, si);
    }
    atomicAdd(&sred[0], sr);   // ds_add_f32
    atomicAdd(&sred[1], si);
    __syncthreads();
    if (threadIdx.x == 0) {
        const float norm  = PIf * ((float)(NTc - 1) * DTf * Ff);
        const float cfac  = DTf * Ff / 0.16f;
        const float scale = (DYf / DTf) * cfac / norm;
        const float ar = sred[0] * scale;
        const float ai = sred[1] * scale;
        out[0] = sqrtf(ar * ar + ai * ai);
    }
}

// ---------------------------------------------------------------------------
extern "C" void kernel_launch(void* const* d_in, const int* in_sizes, int n_in,
                              void* d_out, int out_size, void* d_ws, size_t ws_size,
                              hipStream_t stream) {
    const float* m    = (const float*)d_in[0];   // (8,3,192,192) f32
    const float* msat = (const float*)d_in[1];   // scalar f32
    float* out = (float*)d_out;

    float* W  = (float*)d_ws;           // 16 * 36864 floats (t-rows 8..15 zero)
    float* A1 = W + 16 * NK;            // 8 * 2304 floats

    {   // Kernel 1: curl + folded boundary terms
        const int total = 16 * NK;
        prep_W<<<(total + 255) / 256, 256, 0, stream>>>(m, msat, W);
    }
    {   // Kernel 2: WMMA contraction over k = 36864
        gemm_A1<<<NTILES, 128, 0, stream>>>(W, A1);
    }
    {   // Kernel 3: scalar reduction
        finalize<<<1, 256, 0, stream>>>(A1, out);
    }
}